// ConvGraph_82377472737886
// MI455X (gfx1250) — compile-verified
//
#include <hip/hip_runtime.h>

typedef __attribute__((ext_vector_type(2))) float v2f;
typedef __attribute__((ext_vector_type(8))) float v8f;

#define D_IN  256
#define D_OUT 128

// ---------------------------------------------------------------------------
// Phase 1: h[n, 128] = x[n, 256] @ W[256, 128], fp32 via V_WMMA_F32_16X16X4_F32
// Block: 256 threads = 8 wave32s. Block owns a 16-row M-tile; wave w owns
// N-tile [16w, 16w+16). 64 WMMA ops per wave (K = 256 / 4).
// ---------------------------------------------------------------------------
__global__ void __launch_bounds__(256)
gcn_gemm_wmma_f32(const float* __restrict__ x, const float* __restrict__ W,
                  float* __restrict__ h, int n_nodes) {
  const int lane  = threadIdx.x & 31;
  const int wave  = threadIdx.x >> 5;          // 0..7 -> N tile
  const int m0    = blockIdx.x * 16;
  const int n0    = wave * 16;
  const int lhalf = lane >> 4;                 // 0: lanes 0-15, 1: lanes 16-31
  const int lm    = lane & 15;

  // Clamp A row for ragged tails (keeps EXEC all-ones through the WMMAs).
  int arow_idx = m0 + lm;
  if (arow_idx >= n_nodes) arow_idx = n_nodes - 1;

  // A frag: lanes 0-15 -> row m0+lm, K pair {k0,k0+1}; lanes 16-31 -> {k0+2,k0+3}
  const float* arow = x + (size_t)arow_idx * D_IN + 2 * lhalf;
  // B frag: lanes 0-15 -> col n0+lm, K rows {k0,k0+1}; lanes 16-31 -> {k0+2,k0+3}
  const float* bcol = W + (size_t)(2 * lhalf) * D_OUT + (n0 + lm);

  v8f c = {};
#pragma unroll 4
  for (int k0 = 0; k0 < D_IN; k0 += 4) {
    v2f a = *(const v2f*)(arow + k0);
    v2f b;
    b.x = bcol[(size_t)k0 * D_OUT];
    b.y = bcol[(size_t)k0 * D_OUT + D_OUT];
    // 8-arg form: (neg_a, A, neg_b, B, c_mod, C, reuse_a, reuse_b)
    c = __builtin_amdgcn_wmma_f32_16x16x4_f32(false, a, false, b,
                                              (short)0, c, false, false);
  }

  // C/D layout: VGPR v -> M = m0 + v + 8*lhalf, N = n0 + lm
  float* hbase = h + (size_t)(m0 + 8 * lhalf) * D_OUT + n0 + lm;
  if (m0 + 16 <= n_nodes) {
    // Full tile (always taken for N=100000): unguarded 8-store clause.
#pragma unroll
    for (int v = 0; v < 8; ++v)
      hbase[(size_t)v * D_OUT] = c[v];
  } else {
#pragma unroll
    for (int v = 0; v < 8; ++v)
      if (m0 + v + 8 * lhalf < n_nodes)
        hbase[(size_t)v * D_OUT] = c[v];
  }
}

// ---------------------------------------------------------------------------
// Phase 2: out[dst_e] += w_e * h[src_e]   (E = 3.2M, feature dim 128)
// One wave per edge; edge metadata made wave-uniform (readfirstlane) so it
// takes the scalar-load path. Each lane owns 4 consecutive floats: float4
// gather from L2-resident h, then 4 native fp32 atomic adds (inline asm
// global_atomic_add_f32, device scope) into L2-resident out.
// ---------------------------------------------------------------------------
__device__ __forceinline__ void atomic_add_f32_dev(float* addr, float val) {
  asm volatile("global_atomic_add_f32 %0, %1, off scope:SCOPE_DEV"
               :
               : "v"(addr), "v"(val)
               : "memory");
}

__global__ void __launch_bounds__(256)
gcn_scatter_edges(const float* __restrict__ h, const int* __restrict__ esrc,
                  const int* __restrict__ edst, const float* __restrict__ ew,
                  float* __restrict__ out, int n_edges) {
  const int lane = threadIdx.x & 31;
  int e = blockIdx.x * 8 + (threadIdx.x >> 5);
  if (e >= n_edges) return;
  e = __builtin_amdgcn_readfirstlane(e);   // wave-uniform -> s_load path

  const int   src = esrc[e];
  const int   dst = edst[e];
  const float w   = ew[e];

  const float4 hv = *(const float4*)(h + (size_t)src * D_OUT + lane * 4);
  float* o = out + (size_t)dst * D_OUT + lane * 4;

  atomic_add_f32_dev(o + 0, w * hv.x);
  atomic_add_f32_dev(o + 1, w * hv.y);
  atomic_add_f32_dev(o + 2, w * hv.z);
  atomic_add_f32_dev(o + 3, w * hv.w);
}

// ---------------------------------------------------------------------------
extern "C" void kernel_launch(void* const* d_in, const int* in_sizes, int n_in,
                              void* d_out, int out_size, void* d_ws, size_t ws_size,
                              hipStream_t stream) {
  const float* x    = (const float*)d_in[0];
  const float* W    = (const float*)d_in[1];
  const int*   esrc = (const int*)d_in[2];
  const int*   edst = (const int*)d_in[3];
  const float* ew   = (const float*)d_in[4];
  float*       out  = (float*)d_out;
  float*       h    = (float*)d_ws;   // 100000 * 128 * 4B = 51.2 MB scratch

  const int n_nodes = in_sizes[0] / D_IN;
  const int n_edges = in_sizes[2];

  // Zero the accumulation target (capture-legal async memset).
  hipMemsetAsync(d_out, 0, (size_t)out_size * sizeof(float), stream);

  const int mblocks = (n_nodes + 15) / 16;          // 6250
  gcn_gemm_wmma_f32<<<mblocks, 256, 0, stream>>>(x, W, h, n_nodes);

  const int eblocks = (n_edges + 7) / 8;            // 400000
  gcn_scatter_edges<<<eblocks, 256, 0, stream>>>(h, esrc, edst, ew, out, n_edges);
}